// ConvFCLIFNet_53102975648224
// MI455X (gfx1250) — compile-verified
//
#include <hip/hip_runtime.h>

typedef __attribute__((ext_vector_type(2))) float v2f;
typedef __attribute__((ext_vector_type(8))) float v8f;

#define TT   64
#define BB   512
#define NN   729
#define KDIM 784          // 28*28 pixels = GEMM K after conv folding
#define NPAD 736          // 46 * 16
#define TILE_N 16

// ---------------------------------------------------------------------------
// Kernel 1: fold the 2x2 conv into the FC weight.
// W_effT[p][o] = sum_{dr,dc} conv_w[dr,dc] * fc_w[o, (r-dr)*27 + (c-dc)]
// stored K-major (784 rows x 736 padded cols) for the GEMM B operand.
// ---------------------------------------------------------------------------
__global__ __launch_bounds__(256) void build_weff_kernel(
    const float* __restrict__ fc_w,    // (729, 729) row-major (o, i)
    const float* __restrict__ conv_w,  // (4,)  [[1,2],[3,4]]
    float* __restrict__ weffT)         // (784, 736)
{
    int idx = blockIdx.x * 256 + threadIdx.x;
    if (idx >= KDIM * NPAD) return;
    int p = idx / NPAD;          // pixel index 0..783
    int o = idx % NPAD;          // output neuron (padded)
    float acc = 0.f;
    if (o < NN) {
        int r = p / 28, c = p % 28;
        #pragma unroll
        for (int dr = 0; dr < 2; ++dr) {
            #pragma unroll
            for (int dc = 0; dc < 2; ++dc) {
                int rr = r - dr, cc = c - dc;
                if (rr >= 0 && rr < 27 && cc >= 0 && cc < 27)
                    acc += conv_w[dr * 2 + dc] * fc_w[o * NN + rr * 27 + cc];
            }
        }
    }
    weffT[p * NPAD + o] = acc;
}

// ---------------------------------------------------------------------------
// Kernel 2: fused GEMM (fp32 WMMA 16x16x4) + LIF scan.
// Block = 256 threads (8 wave32). blockIdx.x = n-tile (46), blockIdx.y = b-tile (32).
// Each block computes fc[t, b_tile, n_tile] for ALL t (wave w owns t = 8w..8w+7),
// parks results in LDS, then runs the sequential LIF recurrence in-block.
// ---------------------------------------------------------------------------
__global__ __launch_bounds__(256) void gemm_lif_kernel(
    const float* __restrict__ x,      // (T*B, 784) == x_seq reshaped
    const float* __restrict__ weffT,  // (784, 736)
    const float* __restrict__ fc_b,   // (729,)
    float* __restrict__ out)          // (T, B, 729) spikes
{
    // B panel, swizzled: group g (=K/4): [half(2)][n(16)][kpair(2)] -> ds_load_b64/lane
    __shared__ float Bs[(KDIM / 4) * 64];    // 12544 floats = 49 KB
    __shared__ float Cs[TT * 16 * TILE_N];   // 16384 floats = 64 KB (gfx1250: 320KB LDS)

    const int tid    = threadIdx.x;
    const int lane   = tid & 31;
    const int wave   = tid >> 5;
    const int n_base = blockIdx.x * TILE_N;
    const int b_base = blockIdx.y * 16;

    // --- cooperative swizzled fill of the B panel via CDNA5 async-to-LDS ---
    // Per-lane: copies one float of weffT (row K, column n_base + n) into Bs[idx] in LDS; gather handled by the async
    // unit (ASYNCcnt), no VGPR round trip.
    for (int idx = tid; idx < (KDIM / 4) * 64; idx += 256) {
        int g    = idx >> 6;
        int rem  = idx & 63;
        int half = rem >> 5;          // lane group (0: lanes 0-15, 1: lanes 16-31)
        int n    = (rem & 31) >> 1;
        int kk   = rem & 1;
        int K    = g * 4 + half * 2 + kk;
        const float* gsrc = weffT + K * NPAD + n_base + n;
        unsigned lds_addr = (unsigned)(uintptr_t)(&Bs[idx]);  // low 32b = LDS offset
        asm volatile("global_load_async_to_lds_b32 %0, %1, off"
                     :: "v"(lds_addr), "v"(gsrc) : "memory");
    }
    asm volatile("s_wait_asynccnt 0x0" ::: "memory");
    __syncthreads();

    // WMMA operand addressing (wave32, f32 16x16x4):
    //  A: lane L holds M = L&15, K-pair = (L>>4)*2 .. +1   (2 VGPRs)
    //  B: lane L holds N = L&15, K-pair = (L>>4)*2 .. +1   (2 VGPRs)
    //  C: lane L, VGPR r holds M = r + (L>>4)*8, N = L&15
    const int mlane = lane & 15;
    const int kgrp  = (lane >> 4) * 2;
    const float* bs_lane = Bs + (lane >> 4) * 32 + mlane * 2;

    #pragma unroll 1
    for (int i = 0; i < 8; ++i) {
        const int t = wave * 8 + i;
        const float* arow =
            x + (size_t)(t * BB + b_base + mlane) * KDIM + kgrp;
        if (i < 7) {   // warm L2 for the next timestep's A row (global_prefetch_b8)
            __builtin_prefetch(arow + (size_t)BB * KDIM, 0, 0);
        }
        // Two independent accumulator chains to pipeline the XDL WMMAs.
        v8f c0 = {}, c1 = {};
        #pragma unroll 4
        for (int g = 0; g < KDIM / 4; g += 2) {
            v2f a0 = *(const v2f*)(arow + g * 4);
            v2f b0 = *(const v2f*)(bs_lane + g * 64);
            v2f a1 = *(const v2f*)(arow + (g + 1) * 4);
            v2f b1 = *(const v2f*)(bs_lane + (g + 1) * 64);
            c0 = __builtin_amdgcn_wmma_f32_16x16x4_f32(
                     false, a0, false, b0, (short)0, c0, false, false);
            c1 = __builtin_amdgcn_wmma_f32_16x16x4_f32(
                     false, a1, false, b1, (short)0, c1, false, false);
        }
        // park fc tile in LDS: Cs[t][M][N]
        float* cdst = Cs + t * 256 + ((lane >> 4) * 8) * 16 + mlane;
        #pragma unroll
        for (int r = 0; r < 8; ++r) cdst[r * 16] = c0[r] + c1[r];
    }
    __syncthreads();

    // --- LIF scan: one thread per (b_local, n_local), sequential over t ---
    const int b_local = tid >> 4;
    const int n_local = tid & 15;
    const int n = n_base + n_local;
    const bool valid = (n < NN);
    const float bval = valid ? fc_b[n] : 0.f;
    const float* csrc = Cs + b_local * 16 + n_local;
    const size_t obase = (size_t)(b_base + b_local) * NN + n;

    float v = 0.f;
    #pragma unroll 1
    for (int t = 0; t < TT; ++t) {
        float inp = csrc[t * 256] + bval;
        v = (v + inp) * 0.5f;                       // v += (inp - v)/tau, tau=2
        float s = (v >= 1.0f) ? 1.0f : 0.0f;        // forward heaviside
        if (valid) out[(size_t)t * BB * NN + obase] = s;
        v *= (1.0f - s);                            // hard reset to 0
    }
}

// ---------------------------------------------------------------------------
extern "C" void kernel_launch(void* const* d_in, const int* in_sizes, int n_in,
                              void* d_out, int out_size, void* d_ws, size_t ws_size,
                              hipStream_t stream) {
    const float* x      = (const float*)d_in[0];  // (64,512,1,28,28)
    const float* conv_w = (const float*)d_in[1];  // (1,1,2,2)
    const float* fc_w   = (const float*)d_in[2];  // (729,729)
    const float* fc_b   = (const float*)d_in[3];  // (729,)
    float* out  = (float*)d_out;                  // (64,512,729)
    float* weff = (float*)d_ws;                   // 784*736 floats = 2.31 MB

    build_weff_kernel<<<(KDIM * NPAD + 255) / 256, 256, 0, stream>>>(
        fc_w, conv_w, weff);

    dim3 grid(NPAD / TILE_N, BB / 16);            // (46, 32)
    gemm_lif_kernel<<<grid, 256, 0, stream>>>(x, weff, fc_b, out);
}